// HeteroGNNEncoder_50757923504867
// MI455X (gfx1250) — compile-verified
//
#include <hip/hip_runtime.h>
#include <math.h>

// ---------------------------------------------------------------------------
// Problem constants (match reference)
// ---------------------------------------------------------------------------
#define N_NODES 100000
#define EDGES   1600000
#define HID     128
#define OUTC    64
#define BN_EPS  1e-5f

typedef float v2f __attribute__((ext_vector_type(2)));
typedef float v8f __attribute__((ext_vector_type(8)));

// ---------------------------------------------------------------------------
// float atomic-max via order-preserving uint encoding
// ---------------------------------------------------------------------------
__device__ __forceinline__ unsigned enc_f(float x) {
    unsigned b = __float_as_uint(x);
    return (b & 0x80000000u) ? ~b : (b | 0x80000000u);
}
__device__ __forceinline__ float dec_f(unsigned e) {
    unsigned b = (e & 0x80000000u) ? (e & 0x7FFFFFFFu) : ~e;
    return __uint_as_float(b);
}
__device__ __forceinline__ float lrelu02(float x) { return x >= 0.f ? x : 0.2f * x; }

// ---------------------------------------------------------------------------
// SAGE aggregation scatters (mean aggregation numerator + degree)
// ---------------------------------------------------------------------------
__global__ void scatter_wallet_kernel(const int* __restrict__ src, const int* __restrict__ dst,
                                      const float* __restrict__ xw,
                                      float* __restrict__ agg, float* __restrict__ deg) {
    int e = blockIdx.x * blockDim.x + threadIdx.x;
    if (e >= EDGES) return;
    int s = src[e], d = dst[e];
    const float4 v = *(const float4*)(xw + (size_t)s * 4);
    float* a = agg + (size_t)d * 4;
    atomicAdd(a + 0, v.x); atomicAdd(a + 1, v.y);
    atomicAdd(a + 2, v.z); atomicAdd(a + 3, v.w);
    atomicAdd(&deg[d], 1.f);
}

__global__ void scatter_user_kernel(const int* __restrict__ src, const int* __restrict__ dst,
                                    const float* __restrict__ xu,
                                    float* __restrict__ agg, float* __restrict__ deg) {
    int idx = blockIdx.x * blockDim.x + threadIdx.x;
    if (idx >= EDGES * 4) return;
    int e = idx >> 2, q = idx & 3;
    int s = src[e], d = dst[e];
    const float4 v = *(const float4*)(xu + (size_t)s * 16 + q * 4);
    float* a = agg + (size_t)d * 16 + q * 4;
    atomicAdd(a + 0, v.x); atomicAdd(a + 1, v.y);
    atomicAdd(a + 2, v.z); atomicAdd(a + 3, v.w);
    if (q == 0) atomicAdd(&deg[d], 1.f);
}

// ---------------------------------------------------------------------------
// Fused per-node SAGE MLPs:
//   h = relu(agg_wu/deg @ Wl_wu + b_wu + x_u @ Wr_wu)
//     + relu(agg_uu/deg @ Wl_uu + b_uu + x_u @ Wr_uu)
// One block (128 threads) per node; thread t = output channel t.
// ---------------------------------------------------------------------------
__global__ void node_mlp_kernel(const float* __restrict__ xu,
                                const float* __restrict__ agg_wu, const float* __restrict__ deg_wu,
                                const float* __restrict__ agg_uu, const float* __restrict__ deg_uu,
                                const float* __restrict__ wl_wu, const float* __restrict__ b_wu,
                                const float* __restrict__ wr_wu,
                                const float* __restrict__ wl_uu, const float* __restrict__ b_uu,
                                const float* __restrict__ wr_uu,
                                float* __restrict__ hpre) {
    __shared__ float sx[16], su[16], sw[4];
    const int n = blockIdx.x;
    const int t = threadIdx.x;
    if (t < 16) sx[t] = xu[(size_t)n * 16 + t];
    else if (t < 32) {
        float d = fmaxf(deg_uu[n], 1.f);
        su[t - 16] = agg_uu[(size_t)n * 16 + (t - 16)] / d;
    } else if (t < 36) {
        float d = fmaxf(deg_wu[n], 1.f);
        sw[t - 32] = agg_wu[(size_t)n * 4 + (t - 32)] / d;
    }
    __syncthreads();
    float h1 = b_wu[t];
    float h2 = b_uu[t];
#pragma unroll
    for (int k = 0; k < 4; ++k)  h1 += sw[k] * wl_wu[k * HID + t];
#pragma unroll
    for (int k = 0; k < 16; ++k) h1 += sx[k] * wr_wu[k * HID + t];
#pragma unroll
    for (int k = 0; k < 16; ++k) h2 += su[k] * wl_uu[k * HID + t];
#pragma unroll
    for (int k = 0; k < 16; ++k) h2 += sx[k] * wr_uu[k * HID + t];
    hpre[(size_t)n * HID + t] = fmaxf(h1, 0.f) + fmaxf(h2, 0.f);
}

// ---------------------------------------------------------------------------
// BatchNorm stats (per-channel sum / sumsq), 128 threads = 128 channels
// ---------------------------------------------------------------------------
__global__ void stats_kernel(const float* __restrict__ x,
                             float* __restrict__ sum, float* __restrict__ sumsq) {
    const int t = threadIdx.x;
    float s = 0.f, q = 0.f;
    for (int r = blockIdx.x; r < N_NODES; r += gridDim.x) {
        float v = x[(size_t)r * HID + t];
        s += v; q += v * v;
    }
    atomicAdd(&sum[t], s);
    atomicAdd(&sumsq[t], q);
}

// Fused GAT-output bias + ReLU + BN2 stats (writes y back in place)
__global__ void bias_relu_stats_kernel(float* __restrict__ x, const float* __restrict__ bias,
                                       float* __restrict__ sum, float* __restrict__ sumsq) {
    const int t = threadIdx.x;
    const float b = bias[t];
    float s = 0.f, q = 0.f;
    for (int r = blockIdx.x; r < N_NODES; r += gridDim.x) {
        float v = fmaxf(x[(size_t)r * HID + t] + b, 0.f);
        x[(size_t)r * HID + t] = v;
        s += v; q += v * v;
    }
    atomicAdd(&sum[t], s);
    atomicAdd(&sumsq[t], q);
}

__global__ void bn_final_kernel(const float* __restrict__ sum, const float* __restrict__ sumsq,
                                const float* __restrict__ g, const float* __restrict__ b,
                                float* __restrict__ scale, float* __restrict__ shift) {
    const int t = threadIdx.x;
    const float inv_n = 1.f / (float)N_NODES;
    float mean = sum[t] * inv_n;
    float var  = sumsq[t] * inv_n - mean * mean;
    float inv  = rsqrtf(var + BN_EPS);
    float sc   = g[t] * inv;
    scale[t] = sc;
    shift[t] = b[t] - mean * sc;
}

// ---------------------------------------------------------------------------
// WMMA f32 GEMM #1: xp[100000x128] = BN(hpre) @ W[128x128]
// Wave-per-16x16 tile; 8 waves/block cover N=128; grid = 6250 M-tiles.
// A layout (32b A 16x4): lanes 0-15 <-> M rows, lane half selects K {0,1}/{2,3}.
// ---------------------------------------------------------------------------
__global__ void gemm_xp_kernel(const float* __restrict__ hpre,
                               const float* __restrict__ scale, const float* __restrict__ shift,
                               const float* __restrict__ W,
                               float* __restrict__ xp) {
    const int wave  = threadIdx.x >> 5;        // N tile 0..7
    const int lane  = threadIdx.x & 31;
    const int mbase = blockIdx.x * 16;
    const int nbase = wave * 16;
    const int r     = lane & 15;
    const int t2    = (lane >> 4) << 1;        // 0 or 2 (K sub-pair)
    const float* __restrict__ arow = hpre + (size_t)(mbase + r) * HID;
    v8f acc = {};
    for (int k = 0; k < HID; k += 4) {
        const int ka = k + t2;
        v2f a, b;
        a.x = arow[ka]     * scale[ka]     + shift[ka];
        a.y = arow[ka + 1] * scale[ka + 1] + shift[ka + 1];
        b.x = W[(size_t)ka * HID + nbase + r];
        b.y = W[(size_t)(ka + 1) * HID + nbase + r];
        acc = __builtin_amdgcn_wmma_f32_16x16x4_f32(false, a, false, b, (short)0, acc,
                                                    false, false);
    }
    const int rowoff = (lane >> 4) * 8;        // D: VGPR v -> row v (+8 for hi lanes)
    float* __restrict__ outp = xp + (size_t)(mbase + rowoff) * HID + nbase + r;
#pragma unroll
    for (int v = 0; v < 8; ++v) outp[(size_t)v * HID] = acc[v];
}

// ---------------------------------------------------------------------------
// WMMA f32 GEMM #2: out[100000x64] = relu(BN2(h2) @ Wp[128x64] + pb)
// ---------------------------------------------------------------------------
__global__ void gemm_proj_kernel(const float* __restrict__ h2,
                                 const float* __restrict__ scale, const float* __restrict__ shift,
                                 const float* __restrict__ Wp, const float* __restrict__ pb,
                                 float* __restrict__ out) {
    const int wave  = threadIdx.x >> 5;        // N tile 0..3
    const int lane  = threadIdx.x & 31;
    const int mbase = blockIdx.x * 16;
    const int nbase = wave * 16;
    const int r     = lane & 15;
    const int t2    = (lane >> 4) << 1;
    const float* __restrict__ arow = h2 + (size_t)(mbase + r) * HID;
    v8f acc = {};
    for (int k = 0; k < HID; k += 4) {
        const int ka = k + t2;
        v2f a, b;
        a.x = arow[ka]     * scale[ka]     + shift[ka];
        a.y = arow[ka + 1] * scale[ka + 1] + shift[ka + 1];
        b.x = Wp[(size_t)ka * OUTC + nbase + r];
        b.y = Wp[(size_t)(ka + 1) * OUTC + nbase + r];
        acc = __builtin_amdgcn_wmma_f32_16x16x4_f32(false, a, false, b, (short)0, acc,
                                                    false, false);
    }
    const int rowoff = (lane >> 4) * 8;
    const float bias = pb[nbase + r];
    float* __restrict__ outp = out + (size_t)(mbase + rowoff) * OUTC + nbase + r;
#pragma unroll
    for (int v = 0; v < 8; ++v) outp[(size_t)v * OUTC] = fmaxf(acc[v] + bias, 0.f);
}

// ---------------------------------------------------------------------------
// GAT attention: per-(node,head) alpha-source/dest dots
// ---------------------------------------------------------------------------
__global__ void al_kernel(const float* __restrict__ xp,
                          const float* __restrict__ as, const float* __restrict__ ad,
                          float* __restrict__ al_s, float* __restrict__ al_d) {
    int idx = blockIdx.x * blockDim.x + threadIdx.x;
    if (idx >= N_NODES * 4) return;
    const int node = idx >> 2, h = idx & 3;
    const float* xr  = xp + (size_t)node * HID + h * 32;
    const float* pas = as + h * 32;
    const float* pad = ad + h * 32;
    float ss = 0.f, sd = 0.f;
#pragma unroll 8
    for (int c = 0; c < 32; ++c) { float v = xr[c]; ss += v * pas[c]; sd += v * pad[c]; }
    al_s[idx] = ss;
    al_d[idx] = sd;
}

// Edge passes include the n self-loops at indices [EDGES, EDGES+N_NODES)
__global__ void logit_max_kernel(const int* __restrict__ src, const int* __restrict__ dst,
                                 const float* __restrict__ al_s, const float* __restrict__ al_d,
                                 unsigned* __restrict__ menc) {
    int idx = blockIdx.x * blockDim.x + threadIdx.x;
    if (idx >= (EDGES + N_NODES) * 4) return;
    const int e = idx >> 2, h = idx & 3;
    int s_, d_;
    if (e < EDGES) { s_ = src[e]; d_ = dst[e]; } else { s_ = d_ = e - EDGES; }
    float l = lrelu02(al_s[s_ * 4 + h] + al_d[d_ * 4 + h]);
    atomicMax(&menc[d_ * 4 + h], enc_f(l));
}

__global__ void logit_sum_kernel(const int* __restrict__ src, const int* __restrict__ dst,
                                 const float* __restrict__ al_s, const float* __restrict__ al_d,
                                 const unsigned* __restrict__ menc, float* __restrict__ ssum) {
    int idx = blockIdx.x * blockDim.x + threadIdx.x;
    if (idx >= (EDGES + N_NODES) * 4) return;
    const int e = idx >> 2, h = idx & 3;
    int s_, d_;
    if (e < EDGES) { s_ = src[e]; d_ = dst[e]; } else { s_ = d_ = e - EDGES; }
    float l = lrelu02(al_s[s_ * 4 + h] + al_d[d_ * 4 + h]);
    atomicAdd(&ssum[d_ * 4 + h], expf(l - dec_f(menc[d_ * 4 + h])));
}

// Wave per edge; lane L handles channels [4L,4L+4), head = L/8.
// xp (51 MB) fits in the 192 MB L2, so these random 512B gathers stay on-chip.
__global__ void gat_scatter_kernel(const int* __restrict__ src, const int* __restrict__ dst,
                                   const float* __restrict__ al_s, const float* __restrict__ al_d,
                                   const unsigned* __restrict__ menc, const float* __restrict__ ssum,
                                   const float* __restrict__ xp, float* __restrict__ out) {
    const int lane = threadIdx.x & 31;
    const int e = blockIdx.x * (blockDim.x >> 5) + (threadIdx.x >> 5);
    if (e >= EDGES + N_NODES) return;
    int s_, d_;
    if (e < EDGES) { s_ = src[e]; d_ = dst[e]; } else { s_ = d_ = e - EDGES; }
    const int h = lane >> 3;
    float l = lrelu02(al_s[s_ * 4 + h] + al_d[d_ * 4 + h]);
    float alpha = expf(l - dec_f(menc[d_ * 4 + h])) / (ssum[d_ * 4 + h] + 1e-16f);
    const float4 v = *(const float4*)(xp + (size_t)s_ * HID + lane * 4);
    float* o = out + (size_t)d_ * HID + lane * 4;
    atomicAdd(o + 0, v.x * alpha); atomicAdd(o + 1, v.y * alpha);
    atomicAdd(o + 2, v.z * alpha); atomicAdd(o + 3, v.w * alpha);
}

// ---------------------------------------------------------------------------
// Host-side orchestration
// ---------------------------------------------------------------------------
extern "C" void kernel_launch(void* const* d_in, const int* in_sizes, int n_in,
                              void* d_out, int out_size, void* d_ws, size_t ws_size,
                              hipStream_t stream) {
    (void)in_sizes; (void)n_in; (void)out_size; (void)ws_size;

    const float* x_user   = (const float*)d_in[0];
    const float* x_wallet = (const float*)d_in[1];
    // d_in[2..4]: u2w branch — dead code in the reference (its BN result is discarded)
    const float* w_w2u_l  = (const float*)d_in[5];
    const float* b_w2u    = (const float*)d_in[6];
    const float* w_w2u_r  = (const float*)d_in[7];
    const float* w_u2u_l  = (const float*)d_in[8];
    const float* b_u2u    = (const float*)d_in[9];
    const float* w_u2u_r  = (const float*)d_in[10];
    const float* bn_u_g   = (const float*)d_in[11];
    const float* bn_u_b   = (const float*)d_in[12];
    // d_in[13..14]: bn_w — dead code
    const float* gat_w    = (const float*)d_in[15];
    const float* gat_as   = (const float*)d_in[16];
    const float* gat_ad   = (const float*)d_in[17];
    const float* gat_b    = (const float*)d_in[18];
    const float* bn2_g    = (const float*)d_in[19];
    const float* bn2_b    = (const float*)d_in[20];
    const float* proj_w   = (const float*)d_in[21];
    const float* proj_b   = (const float*)d_in[22];
    // d_in[23]: ei_uw — dead code
    const int* ei_wu_src  = (const int*)d_in[24];
    const int* ei_wu_dst  = ei_wu_src + EDGES;
    const int* ei_uu_src  = (const int*)d_in[25];
    const int* ei_uu_dst  = ei_uu_src + EDGES;

    float* out = (float*)d_out;

    // -------- workspace carve-out (256B aligned, ~169 MB) --------
    char* ws = (char*)d_ws;
    size_t off = 0;
    auto carve = [&](size_t bytes) -> char* {
        char* p = ws + off;
        off += (bytes + 255) & ~(size_t)255;
        return p;
    };
    // aggregation region (zeroed as one contiguous memset)
    char*   agg_base = ws + off;
    float*  agg_wu = (float*)carve(sizeof(float) * (size_t)N_NODES * 4);
    float*  deg_wu = (float*)carve(sizeof(float) * (size_t)N_NODES);
    float*  agg_uu = (float*)carve(sizeof(float) * (size_t)N_NODES * 16);
    float*  deg_uu = (float*)carve(sizeof(float) * (size_t)N_NODES);
    size_t  agg_bytes = (size_t)((char*)ws + off - agg_base);

    float*    hpre = (float*)carve(sizeof(float) * (size_t)N_NODES * HID);
    float*    xp   = (float*)carve(sizeof(float) * (size_t)N_NODES * HID);
    float*    al_s = (float*)carve(sizeof(float) * (size_t)N_NODES * 4);
    float*    al_d = (float*)carve(sizeof(float) * (size_t)N_NODES * 4);
    unsigned* menc = (unsigned*)carve(sizeof(unsigned) * (size_t)N_NODES * 4);
    float*    ssum = (float*)carve(sizeof(float) * (size_t)N_NODES * 4);
    float*    h2   = (float*)carve(sizeof(float) * (size_t)N_NODES * HID);

    char*  red_base = ws + off;
    float* sum1 = (float*)carve(sizeof(float) * HID);
    float* sq1  = (float*)carve(sizeof(float) * HID);
    float* sum2 = (float*)carve(sizeof(float) * HID);
    float* sq2  = (float*)carve(sizeof(float) * HID);
    size_t red_bytes = (size_t)((char*)ws + off - red_base);
    float* scale1 = (float*)carve(sizeof(float) * HID);
    float* shift1 = (float*)carve(sizeof(float) * HID);
    float* scale2 = (float*)carve(sizeof(float) * HID);
    float* shift2 = (float*)carve(sizeof(float) * HID);

    // -------- zero-init (re-done every call: deterministic under graph replay) --------
    hipMemsetAsync(agg_base, 0, agg_bytes, stream);
    hipMemsetAsync(menc, 0, sizeof(unsigned) * (size_t)N_NODES * 4, stream); // enc(-inf) floor
    hipMemsetAsync(ssum, 0, sizeof(float) * (size_t)N_NODES * 4, stream);
    hipMemsetAsync(h2,   0, sizeof(float) * (size_t)N_NODES * HID, stream);
    hipMemsetAsync(red_base, 0, red_bytes, stream);

    const int M_EH = EDGES + N_NODES;          // edges + self-loops = 1,700,000

    // -------- Stage A: SAGE aggregation --------
    scatter_wallet_kernel<<<(EDGES + 255) / 256, 256, 0, stream>>>(
        ei_wu_src, ei_wu_dst, x_wallet, agg_wu, deg_wu);
    scatter_user_kernel<<<(EDGES * 4 + 255) / 256, 256, 0, stream>>>(
        ei_uu_src, ei_uu_dst, x_user, agg_uu, deg_uu);

    // -------- Stage B: fused SAGE MLPs -> pre-BN h_user --------
    node_mlp_kernel<<<N_NODES, HID, 0, stream>>>(
        x_user, agg_wu, deg_wu, agg_uu, deg_uu,
        w_w2u_l, b_w2u, w_w2u_r, w_u2u_l, b_u2u, w_u2u_r, hpre);

    // -------- BN1 stats/finalize (BN applied fused inside GEMM A-load) --------
    stats_kernel<<<512, HID, 0, stream>>>(hpre, sum1, sq1);
    bn_final_kernel<<<1, HID, 0, stream>>>(sum1, sq1, bn_u_g, bn_u_b, scale1, shift1);

    // -------- GAT projection GEMM (WMMA f32 16x16x4, BN1 fused) --------
    gemm_xp_kernel<<<N_NODES / 16, 256, 0, stream>>>(hpre, scale1, shift1, gat_w, xp);

    // -------- GAT attention --------
    al_kernel<<<(N_NODES * 4 + 255) / 256, 256, 0, stream>>>(xp, gat_as, gat_ad, al_s, al_d);
    logit_max_kernel<<<(M_EH * 4 + 255) / 256, 256, 0, stream>>>(
        ei_uu_src, ei_uu_dst, al_s, al_d, menc);
    logit_sum_kernel<<<(M_EH * 4 + 255) / 256, 256, 0, stream>>>(
        ei_uu_src, ei_uu_dst, al_s, al_d, menc, ssum);
    gat_scatter_kernel<<<(M_EH + 7) / 8, 256, 0, stream>>>(
        ei_uu_src, ei_uu_dst, al_s, al_d, menc, ssum, xp, h2);

    // -------- bias + ReLU + BN2 stats (fused), finalize --------
    bias_relu_stats_kernel<<<512, HID, 0, stream>>>(h2, gat_b, sum2, sq2);
    bn_final_kernel<<<1, HID, 0, stream>>>(sum2, sq2, bn2_g, bn2_b, scale2, shift2);

    // -------- output projection GEMM (WMMA f32, BN2 + bias + ReLU fused) --------
    gemm_proj_kernel<<<N_NODES / 16, 128, 0, stream>>>(h2, scale2, shift2, proj_w, proj_b, out);
}